// CRPS_37134287241390
// MI455X (gfx1250) — compile-verified
//
#include <hip/hip_runtime.h>
#include <hip/hip_bf16.h>

// Problem constants (predictions [16,32,121,240] f32, targets [16,121,240] f32)
#define B_DIM   16
#define N_ENS   32
#define H_DIM   121
#define W_DIM   240
#define HWSZ    (H_DIM * W_DIM)          // 29040
#define NPTS    (B_DIM * HWSZ)           // 464640 = 1815 * 256 exactly
#define NBLK    (NPTS / 256)             // 1815

typedef __attribute__((ext_vector_type(2))) float v2f;
typedef __attribute__((ext_vector_type(8))) float v8f;

// ---------------------------------------------------------------------------
// Wave-wide (32 lane) f32 sum using V_WMMA_F32_16X16X4_F32, layout-agnostic:
//   A = all ones (entire 16x4 matrix is 1.0 no matter how slots stripe)
//   B = {v, 0} per lane -> each lane's v lands in one distinct B slot
//   D[m][n] = colsum_n(B); documented C layout: lane n, c[0] = colsum(n)
//   (lanes 16..31 hold duplicates of colsum(n-16))
//   Sum of the 16 column sums == sum of all 32 lane values. EXEC must be all 1s.
// ---------------------------------------------------------------------------
__device__ __forceinline__ float wave_sum32_wmma(float v) {
  v2f a; a[0] = 1.0f; a[1] = 1.0f;
  v2f b; b[0] = v;    b[1] = 0.0f;
  v8f c = {};
  c = __builtin_amdgcn_wmma_f32_16x16x4_f32(
      /*neg_a=*/false, a, /*neg_b=*/false, b,
      /*c_mod=*/(short)0, c, /*reuse_a=*/false, /*reuse_b=*/false);
  float s = c[0];                 // colsum(lane & 15), duplicated across halves
  s += __shfl_xor(s, 8, 32);
  s += __shfl_xor(s, 4, 32);
  s += __shfl_xor(s, 2, 32);
  s += __shfl_xor(s, 1, 32);      // all lanes: total over the wave
  return s;
}

// ---------------------------------------------------------------------------
// Kernel 0: latitude weights w[h] = N_LAT * cos(lat_h) / sum_h cos(lat_h)
// ---------------------------------------------------------------------------
__global__ void crps_weights(float* __restrict__ wlat) {
  __shared__ float sc[128];
  const int h = threadIdx.x;
  float c = 0.0f;
  if (h < H_DIM) {
    float lat = 90.0f - 1.5f * (float)h;
    c = cosf(lat * 0.017453292519943295f);   // deg2rad in f32, like JAX
  }
  sc[h] = c;
  __syncthreads();
  #pragma unroll
  for (int s = 64; s > 0; s >>= 1) {
    if (h < s) sc[h] += sc[h + s];
    __syncthreads();
  }
  float S = sc[0];
  if (h < H_DIM) wlat[h] = (float)H_DIM * (c / S);
}

// ---------------------------------------------------------------------------
// Kernel 1: per-point CRPS (weight, |x-y| mean, 32-elem bitonic sort, coef dot)
// + WMMA wave reduction + LDS block reduction -> one partial per block.
// Grid covers NPTS exactly: no tail, EXEC all-1s at the WMMA.
// ---------------------------------------------------------------------------
__global__ __launch_bounds__(256) void crps_main(const float* __restrict__ pred,
                                                 const float* __restrict__ targ,
                                                 const float* __restrict__ wlat,
                                                 float* __restrict__ partials) {
  const int idx = blockIdx.x * 256 + threadIdx.x;   // < NPTS always
  const int b   = idx / HWSZ;
  const int rem = idx - b * HWSZ;
  const int h   = rem / W_DIM;

  const float wl = wlat[h];
  const float y  = targ[idx] * wl;                  // targets flat == idx

  const float* pb = pred + (size_t)b * ((size_t)N_ENS * HWSZ) + rem;

  // Load 32 members (coalesced across lanes per member), weight, term1.
  float xs[N_ENS];
  float t1 = 0.0f;
  #pragma unroll
  for (int n = 0; n < N_ENS; ++n) {
    float x = __builtin_nontemporal_load(&pb[(size_t)n * HWSZ]) * wl;
    xs[n] = x;
    t1 += fabsf(x - y);
  }

  // Fully unrolled bitonic sort (ascending), stays in VGPRs.
  #pragma unroll
  for (int k = 2; k <= N_ENS; k <<= 1) {
    #pragma unroll
    for (int j = k >> 1; j > 0; j >>= 1) {
      #pragma unroll
      for (int i = 0; i < N_ENS; ++i) {
        int l = i ^ j;
        if (l > i) {
          float a  = xs[i], bb = xs[l];
          float lo = fminf(a, bb), hi = fmaxf(a, bb);
          bool  up = ((i & k) == 0);
          xs[i] = up ? lo : hi;
          xs[l] = up ? hi : lo;
        }
      }
    }
  }

  // term2 = sum_i (2i+1-N)/N^2 * xs_sorted[i]
  float t2 = 0.0f;
  #pragma unroll
  for (int i = 0; i < N_ENS; ++i) {
    const float coef = (2.0f * (float)i + 1.0f - (float)N_ENS) *
                       (1.0f / (float)(N_ENS * N_ENS));
    t2 = fmaf(xs[i], coef, t2);
  }

  float v = t1 * (1.0f / (float)N_ENS) - t2;

  // Wave reduce via WMMA, then combine 8 waves in LDS.
  float ws = wave_sum32_wmma(v);
  __shared__ float smem[8];
  const int lane = threadIdx.x & 31;
  const int wid  = threadIdx.x >> 5;
  if (lane == 0) smem[wid] = ws;
  __syncthreads();
  if (threadIdx.x == 0) {
    float s = 0.0f;
    #pragma unroll
    for (int i = 0; i < 8; ++i) s += smem[i];
    partials[blockIdx.x] = s;
  }
}

// ---------------------------------------------------------------------------
// Kernel 2: reduce NBLK partials -> scalar mean
// ---------------------------------------------------------------------------
__global__ __launch_bounds__(256) void crps_final(const float* __restrict__ partials,
                                                  float* __restrict__ out) {
  float v = 0.0f;
  for (int i = threadIdx.x; i < NBLK; i += 256) v += partials[i];

  float ws = wave_sum32_wmma(v);          // all 256 threads reach here: EXEC full
  __shared__ float smem[8];
  const int lane = threadIdx.x & 31;
  const int wid  = threadIdx.x >> 5;
  if (lane == 0) smem[wid] = ws;
  __syncthreads();
  if (threadIdx.x == 0) {
    float s = 0.0f;
    #pragma unroll
    for (int i = 0; i < 8; ++i) s += smem[i];
    out[0] = s * (1.0f / (float)NPTS);
  }
}

// ---------------------------------------------------------------------------
extern "C" void kernel_launch(void* const* d_in, const int* in_sizes, int n_in,
                              void* d_out, int out_size, void* d_ws, size_t ws_size,
                              hipStream_t stream) {
  const float* pred = (const float*)d_in[0];   // [16,32,121,240] f32
  const float* targ = (const float*)d_in[1];   // [16,121,240]    f32
  float* wlat     = (float*)d_ws;              // 121 floats (padded to 128)
  float* partials = wlat + 128;                // NBLK floats

  crps_weights<<<1, 128, 0, stream>>>(wlat);
  crps_main<<<NBLK, 256, 0, stream>>>(pred, targ, wlat, partials);
  crps_final<<<1, 256, 0, stream>>>(partials, (float*)d_out);
}